// RNN_state_4655744549059
// MI455X (gfx1250) — compile-verified
//
#include <hip/hip_runtime.h>
#include <math.h>

// ---- problem constants (from reference) ----
#define DT_C     0.1f
#define SQDT_C   0.316227766016837933f   // sqrt(0.1)
#define NN       1024                    // state dim N
#define TT       1024                    // timesteps
#define IN_DIM   128
#define OUT_DIM  128
#define ST_DIM   10
#define NTRIALS  64

// ---- CDNA5 WMMA types ----
typedef float v2f __attribute__((ext_vector_type(2)));
typedef float v8f __attribute__((ext_vector_type(8)));

// LDS row stride with +4 pad -> bank = (laneM*4 + c) % 64, conflict-free frag reads
#define APAD 1028
#define UPAD 132

// One 16x16x4 fp32 WMMA: A-frag (2 consecutive floats) from padded LDS row,
// B-frag (2 floats, k-major) from a coalesced column of the pre-transposed weight.
__device__ __forceinline__ v8f wmma_step(v8f acc, const float* __restrict__ aRow,
                                         const float* __restrict__ bCol, int k, int ldb) {
    v2f a = *(const v2f*)(aRow + k);
    v2f b;
    b.x = bCol[(size_t)k * ldb];
    b.y = bCol[(size_t)(k + 1) * ldb];
    return __builtin_amdgcn_wmma_f32_16x16x4_f32(false, a, false, b,
                                                 (short)0, acc, false, false);
}

// ---------------------------------------------------------------------------
// One-time prep: transpose dst[c*rows + r] = src[r*cols + c]
// ---------------------------------------------------------------------------
__global__ void transpose_kernel(const float* __restrict__ src, float* __restrict__ dst,
                                 int rows, int cols) {
    int i = blockIdx.x * 256 + threadIdx.x;
    if (i < rows * cols) {
        int r = i / cols, c = i % cols;
        dst[c * rows + r] = src[i];
    }
}

// zero x_seq[:,0,:] and the tanh(x) staging buffer (x0 == 0 -> tanh == 0)
__global__ void init_kernel(float* __restrict__ x_seq, float* __restrict__ tanhX) {
    int i = blockIdx.x * 256 + threadIdx.x;          // 65536 total
    int m = i >> 10, n = i & 1023;
    x_seq[((size_t)m * (TT + 1)) * NN + n] = 0.0f;
    tanhX[i] = 0.0f;
}

// ---------------------------------------------------------------------------
// Step kernel: x_new = (1-dt)*x + dt*(tanhX@Jt + u@Bt) + sqdt*noise*sig^2
// Grid: 64 blocks x 256 thr (8 waves). Block -> (mb = bid>>4, 4 N-tiles).
// Each tile is computed by TWO waves (split-K halves, deterministic LDS
// reduction); each wave runs FOUR independent WMMA accumulator chains.
// ---------------------------------------------------------------------------
__global__ void step_kernel(const float* __restrict__ inp,   const float* __restrict__ noise,
                            const float* __restrict__ Jt,    const float* __restrict__ Bt,
                            const float* __restrict__ sig,   const float* __restrict__ tanhX,
                            float* __restrict__ x_seq, int t)
{
    extern __shared__ float smem[];      // As[16*APAD] | Us[16*UPAD] | Red[4*256]
    float* As  = smem;
    float* Us  = smem + 16 * APAD;
    float* Red = Us   + 16 * UPAD;

    const int tid  = threadIdx.x;
    const int wid  = tid >> 5;
    const int lane = tid & 31;
    const int mb   = blockIdx.x >> 4;            // 0..3   (16-trial row block)
    const int m0   = mb << 4;
    const int tileInBlk = wid >> 1;              // 0..3
    const int kh   = wid & 1;                    // split-K half
    const int n0   = (((blockIdx.x & 15) << 2) + tileInBlk) << 4;

    // coalesced stage of A-operand rows (tanh of previous state, precomputed)
    for (int i = tid; i < 16 * 1024; i += 256) {
        int r = i >> 10, c = i & 1023;
        As[r * APAD + c] = tanhX[(m0 + r) * NN + c];
    }
    // stage u rows for this timestep
    for (int i = tid; i < 16 * IN_DIM; i += 256) {
        int r = i >> 7, c = i & 127;
        Us[r * UPAD + c] = inp[((size_t)(m0 + r) * TT + t) * IN_DIM + c];
    }
    __syncthreads();

    const int laneM = lane & 15;                 // M (A) / N (B) index within tile
    const int laneH = lane >> 4;                 // K-half select inside a chunk
    const int kOff  = laneH << 1;                // 0 or 2

    const float* aRow = &As[laneM * APAD];
    const float* aU   = &Us[laneM * UPAD];
    const float* bJ   = Jt + n0 + laneM;         // Jt[k][n]: lanes coalesced over n
    const float* bB   = Bt + n0 + laneM;         // Bt[j][n]

    v8f c0 = {}, c1 = {}, c2 = {}, c3 = {};

    // ---- recurrent part: tanh(x) @ J^T, this wave's K half (512) ----
    const int kBase = kh << 9;
    for (int kk = kBase; kk < kBase + 512; kk += 16) {
        c0 = wmma_step(c0, aRow, bJ, kk      + kOff, NN);
        c1 = wmma_step(c1, aRow, bJ, kk + 4  + kOff, NN);
        c2 = wmma_step(c2, aRow, bJ, kk + 8  + kOff, NN);
        c3 = wmma_step(c3, aRow, bJ, kk + 12 + kOff, NN);
    }
    // ---- input part: u @ B^T, this wave's K half (64) ----
    const int kBase2 = kh << 6;
    for (int kk = kBase2; kk < kBase2 + 64; kk += 16) {
        c0 = wmma_step(c0, aU, bB, kk      + kOff, NN);
        c1 = wmma_step(c1, aU, bB, kk + 4  + kOff, NN);
        c2 = wmma_step(c2, aU, bB, kk + 8  + kOff, NN);
        c3 = wmma_step(c3, aU, bB, kk + 12 + kOff, NN);
    }
    v8f acc = (c0 + c1) + (c2 + c3);

    // ---- deterministic split-K reduction through LDS ----
    if (kh) {
        #pragma unroll
        for (int r = 0; r < 8; r++) Red[tileInBlk * 256 + r * 32 + lane] = acc[r];
    }
    __syncthreads();
    if (!kh) {
        #pragma unroll
        for (int r = 0; r < 8; r++) acc[r] += Red[tileInBlk * 256 + r * 32 + lane];

        // epilogue: leaky integration + noise, write x_seq slot t+1
        const int n  = n0 + laneM;
        const float s  = sig[n];
        const float s2 = s * s;
        #pragma unroll
        for (int r = 0; r < 8; r++) {
            int m = m0 + r + (laneH << 3);       // C/D layout: VGPR r, lane-half -> row
            size_t rowBase = ((size_t)m * (TT + 1) + t) * NN;
            float xp = x_seq[rowBase + n];
            float nz = noise[((size_t)m * TT + t) * NN + n];
            float xn = (1.0f - DT_C) * xp + DT_C * acc[r] + SQDT_C * nz * s2;
            x_seq[rowBase + NN + n] = xn;        // == slot (t+1)
        }
    }
}

// ---------------------------------------------------------------------------
// Readout kernel: out = tanh(x_new)@W^T (WMMA), state = onehot(argmax(x_new@A)),
// and stores tanh(x_new) for the NEXT step's A-operand.
// Grid: 4 blocks (one per 16-trial block) x 512 thr (16 waves =
// 8 out-tiles x 2 split-K halves).
// ---------------------------------------------------------------------------
__global__ void readout_kernel(const float* __restrict__ x_seq, const float* __restrict__ Wt,
                               const float* __restrict__ Amat,  float* __restrict__ tanhX,
                               float* __restrict__ out_seq, float* __restrict__ state_seq, int t)
{
    extern __shared__ float smem[];      // Ts[16*APAD] | Red[8*256]
    __shared__ float xa_s[16][ST_DIM];
    float* Ts  = smem;
    float* Red = smem + 16 * APAD;

    const int tid  = threadIdx.x;
    const int wid  = tid >> 5;
    const int lane = tid & 31;
    const int m0   = blockIdx.x << 4;
    const int tileInBlk = wid >> 1;              // 0..7  (16 output cols each)
    const int kh   = wid & 1;

    // compute tanh(x_new) ONCE per element; stage in LDS + persist for next step
    for (int i = tid; i < 16 * 1024; i += 512) {
        int r = i >> 10, c = i & 1023;
        float xv = x_seq[((size_t)(m0 + r) * (TT + 1) + t + 1) * NN + c];
        float th = tanhf(xv);
        Ts[r * APAD + c]         = th;
        tanhX[(m0 + r) * NN + c] = th;
    }
    __syncthreads();

    // ---- out GEMM: tile -> output cols [tile*16, tile*16+16), split-K x2 ----
    const int laneM = lane & 15;
    const int laneH = lane >> 4;
    const int kOff  = laneH << 1;
    const int o0    = tileInBlk << 4;

    const float* aRow = &Ts[laneM * APAD];
    const float* bW   = Wt + o0 + laneM;         // Wt[k][o]: coalesced over o

    v8f c0 = {}, c1 = {}, c2 = {}, c3 = {};
    const int kBase = kh << 9;
    for (int kk = kBase; kk < kBase + 512; kk += 16) {
        c0 = wmma_step(c0, aRow, bW, kk      + kOff, OUT_DIM);
        c1 = wmma_step(c1, aRow, bW, kk + 4  + kOff, OUT_DIM);
        c2 = wmma_step(c2, aRow, bW, kk + 8  + kOff, OUT_DIM);
        c3 = wmma_step(c3, aRow, bW, kk + 12 + kOff, OUT_DIM);
    }
    v8f acc = (c0 + c1) + (c2 + c3);

    if (kh) {
        #pragma unroll
        for (int r = 0; r < 8; r++) Red[tileInBlk * 256 + r * 32 + lane] = acc[r];
    }
    __syncthreads();
    if (!kh) {
        #pragma unroll
        for (int r = 0; r < 8; r++) {
            float v = acc[r] + Red[tileInBlk * 256 + r * 32 + lane];
            int m = m0 + r + (laneH << 3);
            out_seq[((size_t)m * TT + t) * OUT_DIM + o0 + laneM] = v;
        }
    }

    // ---- xa = x_new @ A (64x10), argmax -> one-hot ----
    if (tid < 16 * ST_DIM) {
        int rl = tid / ST_DIM, s = tid % ST_DIM;
        const float* xrow = &x_seq[((size_t)(m0 + rl) * (TT + 1) + t + 1) * NN];
        float dot = 0.0f;
        for (int k = 0; k < NN; k++) dot += xrow[k] * Amat[k * ST_DIM + s];
        xa_s[rl][s] = dot;
    }
    __syncthreads();
    if (tid < 16) {
        float best = xa_s[tid][0]; int bi = 0;
        #pragma unroll
        for (int s = 1; s < ST_DIM; s++) {       // strict > keeps first max (jnp.argmax)
            float v = xa_s[tid][s];
            if (v > best) { best = v; bi = s; }
        }
        float* sp = &state_seq[((size_t)(m0 + tid) * TT + t) * ST_DIM];
        #pragma unroll
        for (int s = 0; s < ST_DIM; s++) sp[s] = (s == bi) ? 1.0f : 0.0f;
    }
}

// ---------------------------------------------------------------------------
extern "C" void kernel_launch(void* const* d_in, const int* in_sizes, int n_in,
                              void* d_out, int out_size, void* d_ws, size_t ws_size,
                              hipStream_t stream) {
    const float* inp   = (const float*)d_in[0];   // (64, 1024, 128)
    const float* noise = (const float*)d_in[1];   // (64, 1024, 1024)
    const float* B     = (const float*)d_in[2];   // (1024, 128)
    const float* J     = (const float*)d_in[3];   // (1024, 1024)
    const float* W     = (const float*)d_in[4];   // (128, 1024)
    const float* A     = (const float*)d_in[5];   // (1024, 10)
    const float* sig   = (const float*)d_in[6];   // (1024,)

    float* x_seq     = (float*)d_out;                                   // (64,1025,1024)
    float* out_seq   = x_seq   + (size_t)NTRIALS * (TT + 1) * NN;       // (64,1024,128)
    float* state_seq = out_seq + (size_t)NTRIALS * TT * OUT_DIM;        // (64,1024,10)

    // workspace: Jt(1M) | Bt(128K) | Wt(128K) | tanhX(64K) floats
    float* ws    = (float*)d_ws;
    float* Jt    = ws;
    float* Bt    = Jt + (size_t)NN * NN;
    float* Wt    = Bt + (size_t)IN_DIM * NN;
    float* tanhX = Wt + (size_t)NN * OUT_DIM;

    // one-time prep (inside capture; deterministic)
    transpose_kernel<<<(NN * NN + 255) / 256, 256, 0, stream>>>(J, Jt, NN, NN);
    transpose_kernel<<<(NN * IN_DIM + 255) / 256, 256, 0, stream>>>(B, Bt, NN, IN_DIM);
    transpose_kernel<<<(OUT_DIM * NN + 255) / 256, 256, 0, stream>>>(W, Wt, OUT_DIM, NN);
    init_kernel<<<(NTRIALS * NN) / 256, 256, 0, stream>>>(x_seq, tanhX);

    const size_t sh1 = (size_t)(16 * APAD + 16 * UPAD + 4 * 256) * sizeof(float); // ~78 KB
    const size_t sh2 = (size_t)(16 * APAD + 8 * 256) * sizeof(float);             // ~72 KB

    for (int t = 0; t < TT; t++) {
        step_kernel<<<64, 256, sh1, stream>>>(inp, noise, Jt, Bt, sig, tanhX, x_seq, t);
        readout_kernel<<<4, 512, sh2, stream>>>(x_seq, Wt, A, tanhX, out_seq, state_seq, t);
    }
}